// scalarGraph_8358006358516
// MI455X (gfx1250) — compile-verified
//
#include <hip/hip_runtime.h>

// Graph divergence: g[c,e] = x[c,i[e]] - x[c,j[e]];
//                   d[c,i[e]] += g; d[c,j[e]] -= g
// MI455X strategy: all buffers L2-resident (192MB L2); minimize L2 request
// count. x is transposed to node-major (64B per node = 1 cacheline), so each
// node gather is 4x global_load_b128 to one line instead of 16 scattered b32
// to 16 lines (4x fewer gather requests). Scatter-adds are native no-return
// global_atomic_add_f32 (STOREcnt-tracked, fire-and-forget). Transposes are
// dense/coalesced and negligible (~26MB total at 23.3TB/s).

typedef int   v4i __attribute__((ext_vector_type(4)));
typedef float v4f __attribute__((ext_vector_type(4)));

#define CHN 16
#define EPT 4  // edges per thread

// ---------------- main edge kernel (node-major layout) ----------------
__global__ __launch_bounds__(256) void edge_div_nm_kernel(
    const float* __restrict__ xt,   // [N][16]
    const int* __restrict__ iInd,
    const int* __restrict__ jInd,
    float* __restrict__ dt,         // [N][16] (pre-zeroed)
    int E) {
    int t = blockIdx.x * blockDim.x + threadIdx.x;
    long e0 = (long)t * EPT;
    if (e0 >= E) return;

    // Stream-ahead prefetch of the index arrays (global_prefetch_b8).
    __builtin_prefetch(iInd + e0 + 4096, 0, 0);
    __builtin_prefetch(jInd + e0 + 4096, 0, 0);

    if (e0 + (EPT - 1) < E) {
        // Coalesced 128-bit non-temporal index loads (streamed once).
        v4i iv = __builtin_nontemporal_load(
            reinterpret_cast<const v4i*>(iInd + e0));
        v4i jv = __builtin_nontemporal_load(
            reinterpret_cast<const v4i*>(jInd + e0));
#pragma unroll
        for (int k = 0; k < EPT; ++k) {
            const long bi = (long)iv[k] * CHN;
            const long bj = (long)jv[k] * CHN;
            const v4f* xi = reinterpret_cast<const v4f*>(xt + bi);
            const v4f* xj = reinterpret_cast<const v4f*>(xt + bj);
            float* di = dt + bi;
            float* dj = dt + bj;
            v4f g0 = xi[0] - xj[0];
            v4f g1 = xi[1] - xj[1];
            v4f g2 = xi[2] - xj[2];
            v4f g3 = xi[3] - xj[3];
#pragma unroll
            for (int r = 0; r < 4; ++r) {
                (void)unsafeAtomicAdd(di + 0 + r, g0[r]);
                (void)unsafeAtomicAdd(di + 4 + r, g1[r]);
                (void)unsafeAtomicAdd(di + 8 + r, g2[r]);
                (void)unsafeAtomicAdd(di + 12 + r, g3[r]);
                (void)unsafeAtomicAdd(dj + 0 + r, -g0[r]);
                (void)unsafeAtomicAdd(dj + 4 + r, -g1[r]);
                (void)unsafeAtomicAdd(dj + 8 + r, -g2[r]);
                (void)unsafeAtomicAdd(dj + 12 + r, -g3[r]);
            }
        }
    } else {
        for (long e = e0; e < E; ++e) {
            const long bi = (long)iInd[e] * CHN;
            const long bj = (long)jInd[e] * CHN;
#pragma unroll
            for (int c = 0; c < CHN; ++c) {
                float g = xt[bi + c] - xt[bj + c];
                (void)unsafeAtomicAdd(dt + bi + c, g);
                (void)unsafeAtomicAdd(dt + bj + c, -g);
            }
        }
    }
}

// -- transpose x [16][N] -> xt [N][16] AND zero dt [N][16] in one pass --
// Channel offsets c*NODES*4 (<= 6MB) fold into signed 24-bit imm offsets.
template <int NODES>
__global__ __launch_bounds__(256) void prep_kernel(
    const float* __restrict__ x, float* __restrict__ xt,
    float* __restrict__ dt) {
    int n = blockIdx.x * blockDim.x + threadIdx.x;
    if (n >= NODES) return;
    const float* xs = x + n;          // 16 coalesced b32 loads, imm offsets
    v4f a, b, c, d;
#pragma unroll
    for (int r = 0; r < 4; ++r) a[r] = xs[(0 + r) * NODES];
#pragma unroll
    for (int r = 0; r < 4; ++r) b[r] = xs[(4 + r) * NODES];
#pragma unroll
    for (int r = 0; r < 4; ++r) c[r] = xs[(8 + r) * NODES];
#pragma unroll
    for (int r = 0; r < 4; ++r) d[r] = xs[(12 + r) * NODES];
    v4f* o = reinterpret_cast<v4f*>(xt + (long)n * CHN);
    o[0] = a; o[1] = b; o[2] = c; o[3] = d;
    v4f z = {0.f, 0.f, 0.f, 0.f};
    v4f* zo = reinterpret_cast<v4f*>(dt + (long)n * CHN);
    zo[0] = z; zo[1] = z; zo[2] = z; zo[3] = z;
}

// -------- transpose back dt [N][16] -> d [16][N] --------
template <int NODES>
__global__ __launch_bounds__(256) void transpose_d_kernel(
    const float* __restrict__ dt, float* __restrict__ d) {
    int n = blockIdx.x * blockDim.x + threadIdx.x;
    if (n >= NODES) return;
    const v4f* s = reinterpret_cast<const v4f*>(dt + (long)n * CHN);
    v4f a = s[0], b = s[1], c = s[2], e = s[3];
    float* o = d + n;                 // 16 coalesced b32 stores, imm offsets
#pragma unroll
    for (int r = 0; r < 4; ++r) o[(0 + r) * NODES] = a[r];
#pragma unroll
    for (int r = 0; r < 4; ++r) o[(4 + r) * NODES] = b[r];
#pragma unroll
    for (int r = 0; r < 4; ++r) o[(8 + r) * NODES] = c[r];
#pragma unroll
    for (int r = 0; r < 4; ++r) o[(12 + r) * NODES] = e[r];
}

// ---------------- zero init (fallback path only) ----------------
__global__ void zero_f4_kernel(float* __restrict__ d, int n) {
    int idx = blockIdx.x * blockDim.x + threadIdx.x;
    int stride = gridDim.x * blockDim.x;
    int n4 = n >> 2;
    v4f* d4 = reinterpret_cast<v4f*>(d);
    v4f z = {0.f, 0.f, 0.f, 0.f};
    for (int i = idx; i < n4; i += stride) d4[i] = z;
    for (int i = (n4 << 2) + idx; i < n; i += stride) d[i] = 0.f;
}

// ---- generic runtime-N fallback (channel-major, direct to d) ----
__global__ __launch_bounds__(256) void edge_div_rt_kernel(
    const float* __restrict__ x,
    const int* __restrict__ iInd,
    const int* __restrict__ jInd,
    float* __restrict__ d,
    int N, int E) {
    int t = blockIdx.x * blockDim.x + threadIdx.x;
    long e0 = (long)t * EPT;
    for (long e = e0; e < E && e < e0 + EPT; ++e) {
        const int ni = iInd[e];
        const int nj = jInd[e];
#pragma unroll
        for (int c = 0; c < CHN; ++c) {
            float g = x[(long)c * N + ni] - x[(long)c * N + nj];
            (void)unsafeAtomicAdd(d + (long)c * N + ni, g);
            (void)unsafeAtomicAdd(d + (long)c * N + nj, -g);
        }
    }
}

extern "C" void kernel_launch(void* const* d_in, const int* in_sizes, int n_in,
                              void* d_out, int out_size, void* d_ws, size_t ws_size,
                              hipStream_t stream) {
    const float* x  = (const float*)d_in[0];
    const int* iInd = (const int*)d_in[1];
    const int* jInd = (const int*)d_in[2];
    float* d        = (float*)d_out;

    const int N = in_sizes[0] / CHN;  // B=1, C=16
    const int E = in_sizes[1];

    const int threads = 256;  // 8 wave32 waves
    const int edge_blocks = ((E + EPT - 1) / EPT + threads - 1) / threads;
    const int node_blocks = (N + threads - 1) / threads;

    const size_t need = (size_t)N * CHN * sizeof(float) * 2;

    if (N == 100000 && ws_size >= need) {
        float* xt = (float*)d_ws;                    // [N][16]
        float* dt = xt + (size_t)N * CHN;            // [N][16]

        // transpose x -> xt and zero dt (one pass)
        prep_kernel<100000><<<node_blocks, threads, 0, stream>>>(x, xt, dt);
        // gather + scatter-add in node-major layout
        edge_div_nm_kernel<<<edge_blocks, threads, 0, stream>>>(
            xt, iInd, jInd, dt, E);
        // transpose back (writes every element of d_out; no pre-zero needed)
        transpose_d_kernel<100000><<<node_blocks, threads, 0, stream>>>(dt, d);
    } else {
        // fallback: zero output, accumulate channel-major directly
        {
            int blocks = (out_size / 4 + threads - 1) / threads;
            if (blocks < 1) blocks = 1;
            if (blocks > 16384) blocks = 16384;
            zero_f4_kernel<<<blocks, threads, 0, stream>>>(d, out_size);
        }
        edge_div_rt_kernel<<<edge_blocks, threads, 0, stream>>>(
            x, iInd, jInd, d, N, E);
    }
}